// PhysicsPriorExtractor_53068615909517
// MI455X (gfx1250) — compile-verified
//
#include <hip/hip_runtime.h>
#include <cstdint>

typedef __attribute__((ext_vector_type(2))) float v2f;
typedef __attribute__((ext_vector_type(8))) float v8f;

#define B_TOT    32768
#define T_SEQ    64
#define C_IN     30     // K*D = 10*3
#define C_PAD    32
#define C_OUT    61     // 30 spatial + 30 vel + 1 mzeni
#define ROWS_PB  8
#define THREADS  256

#define LN_Q   (-1.2039728043259361f)   /* ln(1 - 0.7) */
#define ALPHA_F 0.7f

// Low 32 bits of a flat shared-aperture address == LDS byte offset
__device__ __forceinline__ unsigned lds_off32(const void* p) {
  return (unsigned)(unsigned long long)p;
}

// CDNA5 async global -> LDS copy, 16B per active lane (ASYNCcnt tracked)
__device__ __forceinline__ void async_ld16(unsigned lds, const void* g) {
  asm volatile("global_load_async_to_lds_b128 %0, %1, off"
               :: "v"(lds), "v"(g) : "memory");
}

__device__ __forceinline__ void wait_async0() {
  asm volatile("s_wait_asynccnt 0" ::: "memory");
}

// Forward-EMA operator entry F[i][j]
__device__ __forceinline__ float fcoef(int i, int j) {
  if (j == 0) return __expf((float)i * LN_Q);          // (1-a)^i
  if (j > i)  return 0.0f;
  return ALPHA_F * __expf((float)(i - j) * LN_Q);      // a*(1-a)^(i-j)
}

__device__ __forceinline__ void issue_row(const float* pr, const float* dr,
                                          float* spStage, float* dtStage, int tid) {
  unsigned spl = lds_off32(spStage);
  unsigned dtl = lds_off32(dtStage);
  // spatial row = 64*30 floats = 7680 B = 480 x b128 chunks; dt row = 256 B = 16 chunks
  async_ld16(spl + (unsigned)tid * 16u, (const char*)pr + tid * 16);
  int c2 = tid + 256;
  if (c2 < 480) {
    async_ld16(spl + (unsigned)c2 * 16u, (const char*)pr + c2 * 16);
  } else if (c2 < 496) {
    int d = c2 - 480;
    async_ld16(dtl + (unsigned)d * 16u, (const char*)dr + d * 16);
  }
}

__global__ __launch_bounds__(THREADS)
void physics_prior_kernel(const float* __restrict__ pose,
                          const float* __restrict__ dts,
                          float* __restrict__ out) {
  __shared__ float S[T_SEQ * T_SEQ];                         // 16 KB smoothing matrix
  __shared__ __align__(16) float spStage[2][T_SEQ * C_IN];   // double-buffered stage
  __shared__ __align__(16) float dtStage[2][T_SEQ];
  __shared__ float V[T_SEQ * C_PAD];                         // velocity (B matrix)
  __shared__ float O[T_SEQ * C_PAD];                         // smoothed velocity
  __shared__ float Mz[T_SEQ];

  const int tid   = threadIdx.x;
  const int wave  = tid >> 5;
  const int lane  = tid & 31;
  const int half  = lane >> 4;   // lane half selects K pair / M half
  const int l16   = lane & 15;
  const int tileM = wave & 3;    // time tile 0..3
  const int tileN = wave >> 2;   // channel tile 0..1

  const long long row0 = (long long)blockIdx.x * ROWS_PB;

  // kick off async staging of the first row, then build S while it flies
  issue_row(pose + row0 * (T_SEQ * C_IN), dts + row0 * T_SEQ,
            &spStage[0][0], &dtStage[0][0], tid);

  for (int idx = tid; idx < T_SEQ * T_SEQ; idx += THREADS) {
    int i = idx >> 6, j = idx & 63;
    S[idx] = 0.5f * (fcoef(i, j) + fcoef(63 - i, 63 - j));   // 0.5*(F + J F J)
  }

  for (int it = 0; it < ROWS_PB; ++it) {
    const int buf = it & 1;
    const long long row = row0 + it;

    wait_async0();        // my wave's async copies for row `it` have landed
    __syncthreads();      // => everyone's have landed; prior reads of buf^1 done

    if (it + 1 < ROWS_PB) {
      issue_row(pose + (row + 1) * (T_SEQ * C_IN), dts + (row + 1) * T_SEQ,
                &spStage[buf ^ 1][0], &dtStage[buf ^ 1][0], tid);
    }

    const float* sp = &spStage[buf][0];
    const float* dt = &dtStage[buf][0];

    // velocity V[t][c] = (sp[t]-sp[t-1]) / clip(dt[t], 1e-6, 0.1); row0 & pad cols = 0
    for (int idx = tid; idx < T_SEQ * C_PAD; idx += THREADS) {
      int t = idx >> 5, c = idx & 31;
      float v = 0.0f;
      if (c < C_IN && t > 0) {
        float d = dt[t];
        d = fminf(fmaxf(d, 1e-6f), 0.1f);
        v = (sp[t * C_IN + c] - sp[(t - 1) * C_IN + c]) / d;
      }
      V[idx] = v;
    }

    // mzeni: mean pelvis velocity telescopes to (pel[63]-pel[0])/63
    if (tid < T_SEQ) {
      float fx = (sp[63 * C_IN + 0] - sp[0]) * (1.0f / 63.0f);
      float fy = (sp[63 * C_IN + 1] - sp[1]) * (1.0f / 63.0f);
      float fz = (sp[63 * C_IN + 2] - sp[2]) * (1.0f / 63.0f);
      float n = sqrtf(fx * fx + fy * fy + fz * fz) + 1e-6f;
      fx /= n; fy /= n; fz /= n;
      int t = tid;
      float dx = sp[t * C_IN +  9] - sp[t * C_IN + 18];   // LANK - RANK, ch 9..11 vs 18..20
      float dy = sp[t * C_IN + 10] - sp[t * C_IN + 19];
      float dz = sp[t * C_IN + 11] - sp[t * C_IN + 20];
      Mz[t] = dx * fx + dy * fy + dz * fz + 1e-6f;
    }
    __syncthreads();

    // O = S(64x64) @ V(64x32) via V_WMMA_F32_16X16X4_F32, one 16x16 tile per wave
    {
      v8f acc = {0.f, 0.f, 0.f, 0.f, 0.f, 0.f, 0.f, 0.f};
      const int m  = tileM * 16 + l16;
      const int n  = tileN * 16 + l16;
      const int kh = 2 * half;          // A/B: VGPR v <-> K = v (+2 for upper lanes)
#pragma unroll
      for (int k0 = 0; k0 < T_SEQ; k0 += 4) {
        v2f a, b;
        a.x = S[m * T_SEQ + (k0 + kh)];
        a.y = S[m * T_SEQ + (k0 + kh + 1)];
        b.x = V[(k0 + kh) * C_PAD + n];
        b.y = V[(k0 + kh + 1) * C_PAD + n];
        acc = __builtin_amdgcn_wmma_f32_16x16x4_f32(
            false, a, false, b, (short)0, acc, false, false);
      }
#pragma unroll
      for (int vi = 0; vi < 8; ++vi) {  // C/D: VGPR vi <-> row M = vi (+8 upper lanes)
        O[(tileM * 16 + vi + 8 * half) * C_PAD + n] = acc[vi];
      }
    }
    __syncthreads();

    // coalesced store: [spatial(30) | smoothed vel(30) | mzeni(1)] per timestep
    float* orow = out + row * (T_SEQ * C_OUT);
    for (int idx = tid; idx < T_SEQ * C_OUT; idx += THREADS) {
      int t = idx / C_OUT;
      int c = idx - t * C_OUT;
      float v;
      if (c < C_IN)          v = sp[t * C_IN + c];
      else if (c < 2 * C_IN) v = O[t * C_PAD + (c - C_IN)];
      else                   v = Mz[t];
      orow[idx] = v;
    }
    __syncthreads();   // stage buffer buf is refilled by the next iteration's prefetch
  }
}

extern "C" void kernel_launch(void* const* d_in, const int* in_sizes, int n_in,
                              void* d_out, int out_size, void* d_ws, size_t ws_size,
                              hipStream_t stream) {
  const float* pose = (const float*)d_in[0];   // (B, T, K, D) f32
  const float* dts  = (const float*)d_in[1];   // (B, T, 1)   f32
  float* out = (float*)d_out;                  // (B, T, 61)  f32
  dim3 grid(B_TOT / ROWS_PB);
  physics_prior_kernel<<<grid, THREADS, 0, stream>>>(pose, dts, out);
}